// Capsule_71021579207255
// MI455X (gfx1250) — compile-verified
//
#include <hip/hip_runtime.h>
#include <math.h>

typedef __attribute__((ext_vector_type(2))) float v2f;
typedef __attribute__((ext_vector_type(8))) float v8f;

#define BATCH    64
#define NCAP     32
#define INCAPS   2048
#define KDIM     64
#define DDIM     64
#define RROWS    64      // input rows per capsule = INCAPS/NCAP
#define ROUTINGS 3
#define CAP_EPS  1e-7f

// ---- LDS layout (float offsets), total 79872 floats = 312 KB ----
#define OFF_BL 0                          // b logits   [32][2048]
#define OFF_MX (OFF_BL + NCAP * INCAPS)   // col max    [2048]
#define OFF_DN (OFF_MX + INCAPS)          // 1/denom    [2048]
#define OFF_E  (OFF_DN + INCAPS)          // exchange   [8192]  (u: [n][cc][kk8], V: [cc][kk8][n])
#define OFF_O  (OFF_E + 8192)             // O accum    [32][64]
#define LDS_FLOATS (OFF_O + NCAP * DDIM)  // 79872

__device__ __forceinline__ v8f wmma_f32(v2f a, v2f b, v8f c) {
  // D = A(16x4 f32) * B(4x16 f32) + C(16x16 f32)
  return __builtin_amdgcn_wmma_f32_16x16x4_f32(false, a, false, b, (short)0, c,
                                               false, false);
}

__device__ __forceinline__ v8f vzero8() {
  v8f z;
#pragma unroll
  for (int i = 0; i < 8; ++i) z[i] = 0.f;
  return z;
}

__global__ __launch_bounds__(1024) void capsule_routing_kernel(
    const float* __restrict__ X,   // [64][2048][64]
    const float* __restrict__ W,   // [64][2048]
    float* __restrict__ out)       // [64][32][64]
{
  extern __shared__ float lds[];
  const int b    = blockIdx.x;
  const int tid  = threadIdx.x;
  const int lane = tid & 31;
  const int wv   = tid >> 5;    // wave id == capsule n == owned W column block cc
  const int l16  = lane & 15;
  const int hi   = lane >> 4;   // lane half
  const int k2   = hi << 1;     // K-offset within a 4-step: 0 or 2

  const float* Xb = X + (size_t)b * INCAPS * KDIM;     // batch slice [2048][64]
  const float* Xn = Xb + (size_t)wv * RROWS * KDIM;    // this capsule's 64x64 slice

  // Warm caches for this wave's private tiles (global_prefetch_b8).
  for (int r = lane; r < RROWS; r += 32) __builtin_prefetch(Xn + r * KDIM, 0, 3);
  for (int k = lane; k < KDIM; k += 32)
    __builtin_prefetch(W + (size_t)k * INCAPS + wv * DDIM, 0, 3);

  // b logits = 0
  for (int i = tid; i < NCAP * INCAPS; i += blockDim.x) lds[OFF_BL + i] = 0.f;
  __syncthreads();

  float o0 = 0.f, o1 = 0.f;

  for (int it = 0; it < ROUTINGS; ++it) {
    // ---------- softmax over capsule axis + clear O accumulator ----------
    for (int i = tid; i < INCAPS; i += blockDim.x) {
      float m = lds[OFF_BL + i];
      for (int n = 1; n < NCAP; ++n) m = fmaxf(m, lds[OFF_BL + n * INCAPS + i]);
      float s = 0.f;
      for (int n = 0; n < NCAP; ++n) s += __expf(lds[OFF_BL + n * INCAPS + i] - m);
      lds[OFF_MX + i] = m;
      lds[OFF_DN + i] = 1.f / s;
    }
    for (int i = tid; i < NCAP * DDIM; i += blockDim.x) lds[OFF_O + i] = 0.f;
    __syncthreads();

    // ---------- O-phase: O[n,d] = sum_cc,k u[n,cc,k] * W[k, cc*64+d] ----------
    // u[cc,k] = sum_r c(n, r*32+cc) * Xn[r,k], computed on the fly one 16-wide
    // k-tile at a time (16 floats live); d-dim split in two passes so the
    // O-partial accumulator is only [2][2] (32 floats live).
    for (int dj = 0; dj < 2; ++dj) {
      v8f ao2[2][2];                    // [n-tile][d-tile(local)]
#pragma unroll
      for (int mt = 0; mt < 2; ++mt)
#pragma unroll
        for (int nl = 0; nl < 2; ++nl) ao2[mt][nl] = vzero8();

      v8f ucur[2] = {vzero8(), vzero8()};   // u[:, ntU*16 .. +16), [cc-tile]

      for (int ck = 0; ck < 8; ++ck) {
        if ((ck & 1) == 0) {
          // GEMM1 tile: u[:, ntU] over K = r = 64
          const int ntU = ck >> 1;
          ucur[0] = vzero8();
          ucur[1] = vzero8();
          for (int kt = 0; kt < 16; ++kt) {
            const int r0 = 4 * kt + k2;
            v2f a[2];
#pragma unroll
            for (int mt = 0; mt < 2; ++mt) {
              const int cc = mt * 16 + l16;
              const int i0 = r0 * 32 + cc;
              const int i1 = i0 + 32;
              a[mt].x = __expf(lds[OFF_BL + wv * INCAPS + i0] - lds[OFF_MX + i0]) * lds[OFF_DN + i0];
              a[mt].y = __expf(lds[OFF_BL + wv * INCAPS + i1] - lds[OFF_MX + i1]) * lds[OFF_DN + i1];
            }
            const float* p = Xn + (size_t)r0 * KDIM + ntU * 16 + l16;
            v2f bb; bb.x = p[0]; bb.y = p[KDIM];
#pragma unroll
            for (int mt = 0; mt < 2; ++mt) ucur[mt] = wmma_f32(a[mt], bb, ucur[mt]);
          }
        }
        // publish u chunk: k in [8ck, 8ck+8)  -> E[n][cc][kk]
        if ((l16 >> 3) == (ck & 1)) {
          const int kk = l16 & 7;
#pragma unroll
          for (int mt = 0; mt < 2; ++mt) {
            float* ep = &lds[OFF_E + wv * 256 + (mt * 16 + hi * 8) * 8 + kk];
#pragma unroll
            for (int v = 0; v < 8; ++v) ep[v * 8] = ucur[mt][v];
          }
        }
        __syncthreads();
#pragma unroll
        for (int kt = 0; kt < 2; ++kt) {
          const int kk = 4 * kt + k2;
          v2f a[2];
#pragma unroll
          for (int mt = 0; mt < 2; ++mt) {
            const int n = mt * 16 + l16;
            a[mt].x = lds[OFF_E + n * 256 + wv * 8 + kk];
            a[mt].y = lds[OFF_E + n * 256 + wv * 8 + kk + 1];
          }
          const int kg = 8 * ck + kk;
#pragma unroll
          for (int nl = 0; nl < 2; ++nl) {
            const int nt = dj * 2 + nl;
            const float* p = W + (size_t)kg * INCAPS + wv * DDIM + nt * 16 + l16;
            v2f bb; bb.x = p[0]; bb.y = p[INCAPS];
#pragma unroll
            for (int mt = 0; mt < 2; ++mt) ao2[mt][nl] = wmma_f32(a[mt], bb, ao2[mt][nl]);
          }
        }
        __syncthreads();
      }
      // reduce this wave's cc=wv contribution into the O accumulator:
      // single base address per tile + constant DS-immediate offsets (v*DDIM).
#pragma unroll
      for (int mt = 0; mt < 2; ++mt)
#pragma unroll
        for (int nl = 0; nl < 2; ++nl) {
          const int d = (dj * 2 + nl) * 16 + l16;
          float* obase = &lds[OFF_O + (mt * 16 + hi * 8) * DDIM + d];
#pragma unroll
          for (int v = 0; v < 8; ++v) atomicAdd(obase + v * DDIM, ao2[mt][nl][v]);
        }
    }
    __syncthreads();

    // ---------- squash (wave n owns row n; wave32 shuffle reduction) ----------
    {
      float x0 = lds[OFF_O + wv * DDIM + lane];
      float x1 = lds[OFF_O + wv * DDIM + 32 + lane];
      float p = x0 * x0 + x1 * x1;
      for (int off = 16; off > 0; off >>= 1) p += __shfl_xor(p, off, 32);
      const float s = p + CAP_EPS;
      const float f = sqrtf(s) / (1.f + s);
      o0 = f * x0; o1 = f * x1;
      lds[OFF_O + wv * DDIM + lane]      = o0;
      lds[OFF_O + wv * DDIM + 32 + lane] = o1;
    }
    __syncthreads();

    if (it == ROUTINGS - 1) {
      out[(size_t)b * (NCAP * DDIM) + wv * DDIM + lane]      = o0;
      out[(size_t)b * (NCAP * DDIM) + wv * DDIM + 32 + lane] = o1;
      break;  // uniform across block
    }

    // ---------- agreement update ----------
    // V[n,cc,k] = sum_d W[k, cc*64+d] * o[n,d]   (wave w owns cc=w; Vw = W_cw @ O^T)
    // db[r,cc]  = sum_k Xn[r,k] * V[n,cc,k]
    // Vw computed on the fly one 16-row k-tile at a time (16 floats live);
    // db r-dim split into two passes so its accumulator is [2][2] (32 floats).
    for (int rj = 0; rj < 2; ++rj) {
      v8f adb2[2][2];                  // [r-tile(local)][cc-tile]
#pragma unroll
      for (int ml = 0; ml < 2; ++ml)
#pragma unroll
        for (int nt = 0; nt < 2; ++nt) adb2[ml][nt] = vzero8();

      v8f vcur[2] = {vzero8(), vzero8()};   // Vw[mtV*16.. , n-tile]

      for (int ck = 0; ck < 8; ++ck) {
        if ((ck & 1) == 0) {
          // V tile: Vw[k-tile mtV, n] over K = d = 64
          const int mtV = ck >> 1;
          vcur[0] = vzero8();
          vcur[1] = vzero8();
          for (int kt = 0; kt < 16; ++kt) {
            const int d0 = 4 * kt + k2;
            const float* p = W + (size_t)(mtV * 16 + l16) * INCAPS + wv * DDIM + d0;
            v2f a; a.x = p[0]; a.y = p[1];
#pragma unroll
            for (int nt = 0; nt < 2; ++nt) {
              const int n = nt * 16 + l16;
              v2f bb;
              bb.x = lds[OFF_O + n * DDIM + d0];
              bb.y = lds[OFF_O + n * DDIM + d0 + 1];
              vcur[nt] = wmma_f32(a, bb, vcur[nt]);
            }
          }
        }
        // publish V chunk: k in [8ck, 8ck+8)  -> E2[cc][kk][n]
        if (hi == (ck & 1)) {
#pragma unroll
          for (int nt = 0; nt < 2; ++nt) {
            float* ep = &lds[OFF_E + wv * 256 + nt * 16 + l16];
#pragma unroll
            for (int v = 0; v < 8; ++v) ep[v * 32] = vcur[nt][v];
          }
        }
        __syncthreads();
#pragma unroll
        for (int kt = 0; kt < 2; ++kt) {
          const int kk = 4 * kt + k2;
          const int kg = 8 * ck + kk;
          v2f a[2];
#pragma unroll
          for (int ml = 0; ml < 2; ++ml) {
            const int r = (rj * 2 + ml) * 16 + l16;
            const float* p = Xn + (size_t)r * KDIM + kg;
            a[ml].x = p[0]; a[ml].y = p[1];
          }
#pragma unroll
          for (int nt = 0; nt < 2; ++nt) {
            v2f bb;
            bb.x = lds[OFF_E + (nt * 16 + l16) * 256 + kk * 32 + wv];
            bb.y = lds[OFF_E + (nt * 16 + l16) * 256 + (kk + 1) * 32 + wv];
#pragma unroll
            for (int ml = 0; ml < 2; ++ml) adb2[ml][nt] = wmma_f32(a[ml], bb, adb2[ml][nt]);
          }
        }
        __syncthreads();
      }
      // b[n, r*32+cc] += db: single base per tile + constant offsets (v*32).
#pragma unroll
      for (int ml = 0; ml < 2; ++ml)
#pragma unroll
        for (int nt = 0; nt < 2; ++nt) {
          float* bbase = &lds[OFF_BL + wv * INCAPS +
                              ((rj * 2 + ml) * 16 + hi * 8) * 32 + nt * 16 + l16];
#pragma unroll
          for (int v = 0; v < 8; ++v) bbase[v * 32] += adb2[ml][nt][v];
        }
    }
    __syncthreads();
  }
}

extern "C" void kernel_launch(void* const* d_in, const int* in_sizes, int n_in,
                              void* d_out, int out_size, void* d_ws, size_t ws_size,
                              hipStream_t stream) {
  const float* X = (const float*)d_in[0];   // [64][2048][64]
  const float* W = (const float*)d_in[1];   // [1][64][2048]
  float* out = (float*)d_out;               // [64][32][64]
  (void)in_sizes; (void)n_in; (void)out_size; (void)d_ws; (void)ws_size;

  const size_t shmem = (size_t)LDS_FLOATS * sizeof(float);  // 312 KB of the 320 KB WGP LDS
  hipFuncSetAttribute((const void*)capsule_routing_kernel,
                      hipFuncAttributeMaxDynamicSharedMemorySize, (int)shmem);
  capsule_routing_kernel<<<BATCH, 1024, shmem, stream>>>(X, W, out);
}